// Render_13554916786339
// MI455X (gfx1250) — compile-verified
//
#include <hip/hip_runtime.h>

// CDNA5 / gfx1250 rasterizer.
// Per-pixel winner = last covering triangle attaining max interpolated z.
// Edge functions + z are affine in (px,py) -> evaluated with V_WMMA_F32_16X16X4_F32:
//   A (16x4) = 4 triangles x 4 functions (pAB,pCB,pCA,z) coefficients [a,b,c,0]
//   B (4x16) = 16 pixels, rows [px, py, 1, 0]
// D layout (ISA 7.12.2): lane l<16 holds pixel l rows 0..7 (2 tris); lane l+16 the
// same pixel rows 8..15 (2 more tris). Per-lane argmax, merged via shfl_xor(16).
// Coefficient/bounds table staged into LDS with TENSOR_LOAD_TO_LDS (TDM);
// winner RGBA recomputed exactly in reference op order.

typedef float v2f __attribute__((ext_vector_type(2)));
typedef float v8f __attribute__((ext_vector_type(8)));
typedef unsigned int v4u __attribute__((ext_vector_type(4)));
typedef int v4i __attribute__((ext_vector_type(4)));
typedef int v8i __attribute__((ext_vector_type(8)));

#define SIZE 512
#define NTRI 512
#define COEF_OFF 64                  // floats: wsf[0] = zmin, coefs at wsf+64
#define BND_OFF  (COEF_OFF + NTRI*16)
#define TAB_DW   (NTRI*16 + NTRI*4)  // 10240 dwords = 40KB (coefs then bounds)

__global__ __launch_bounds__(512) void prep_kernel(const float* __restrict__ tris,
                                                   float* __restrict__ wsf) {
    __shared__ float red[512];
    int t = threadIdx.x;
    const float* tp = tris + t * 9;
    float ax = tp[0], ay = tp[1], az = tp[2];
    float bx = tp[3], by = tp[4], bz = tp[5];
    float cx = tp[6], cy = tp[7], cz = tp[8];

    float w = (bx - ax) * (cy - ay) - (by - ay) * (cx - ax);
    bool valid = (w >= 1e-9f);
    float ws_ = valid ? w : 1.0f;

    // area2d(a, p, c) = px*(c.y-a.y) - py*(c.x-a.x) + [a.y*(c.x-a.x) - a.x*(c.y-a.y)]
    float aAB = ay - by, bAB = bx - ax, cAB = by * (ax - bx) - bx * (ay - by);   // area2d(B,p,A)
    float aCB = by - cy, bCB = cx - bx, cCB = cy * (bx - cx) - cx * (by - cy);   // area2d(C,p,B)
    float aCA = cy - ay, bCA = ax - cx, cCA = ay * (cx - ax) - ax * (cy - ay);   // area2d(A,p,C)
    if (!valid) { aAB = 0.0f; bAB = 0.0f; cAB = -1.0f; }  // never covers

    // z = cz + (pCB*(az-cz) + pCA*(bz-cz)) / ws  -> affine coefficients
    float u = (az - cz) / ws_;
    float v = (bz - cz) / ws_;
    float azc = aCB * u + aCA * v;
    float bzc = bCB * u + bCA * v;
    float czc = cCB * u + cCA * v + cz;

    float* cb = wsf + COEF_OFF + t * 16;
    cb[0]  = aAB; cb[1]  = bAB; cb[2]  = cAB; cb[3]  = 0.0f;
    cb[4]  = aCB; cb[5]  = bCB; cb[6]  = cCB; cb[7]  = 0.0f;
    cb[8]  = aCA; cb[9]  = bCA; cb[10] = cCA; cb[11] = 0.0f;
    cb[12] = azc; cb[13] = bzc; cb[14] = czc; cb[15] = 0.0f;

    // AABB -> integer pixel bounds (half = 256)
    float xmn = fminf(ax, fminf(bx, cx)), ymn = fminf(ay, fminf(by, cy));
    float xmx = fmaxf(ax, fmaxf(bx, cx)), ymx = fmaxf(ay, fmaxf(by, cy));
    int* ib = (int*)(wsf + BND_OFF) + t * 4;
    ib[0] = (int)floorf((fminf(fmaxf(xmn, -1.0f), 1.0f) + 1.0f) * 256.0f);
    ib[1] = (int)floorf((fminf(fmaxf(ymn, -1.0f), 1.0f) + 1.0f) * 256.0f);
    ib[2] = (int)floorf((fminf(fmaxf(xmx, -1.0f), 1.0f) + 1.0f) * 256.0f);
    ib[3] = (int)floorf((fminf(fmaxf(ymx, -1.0f), 1.0f) + 1.0f) * 256.0f);

    // global min vertex z (z-buffer seed)
    red[t] = fminf(az, fminf(bz, cz));
    __syncthreads();
    for (int s = 256; s > 0; s >>= 1) {
        if (t < s) red[t] = fminf(red[t], red[t + s]);
        __syncthreads();
    }
    if (t == 0) wsf[0] = red[0];
}

__global__ __launch_bounds__(256) void raster_kernel(const float* __restrict__ tris,
                                                     const float* __restrict__ wsf,
                                                     float* __restrict__ out) {
    __shared__ unsigned int sTab[TAB_DW];  // 40KB: 8192 coef dwords + 2048 bound dwords
    float* sCoef = (float*)sTab;
    int*   sBnd  = (int*)(sTab + NTRI * 16);
    int tid = threadIdx.x;
    float zmin = wsf[0];

#if __has_builtin(__builtin_amdgcn_tensor_load_to_lds)
    // Tensor Data Mover: one DMA per workgroup stages the 40KB table into LDS.
    if (tid == 0) {
        unsigned long long ga = (unsigned long long)(const void*)(wsf + COEF_OFF);
        unsigned int lds = (unsigned int)(uintptr_t)(void*)sTab;  // addrspace(3) offset
        v4u g0;
        g0.x = 1u;                                              // count=1 (user D#)
        g0.y = lds;                                             // lds_addr
        g0.z = (unsigned int)(ga & 0xFFFFFFFFull);              // global_addr[31:0]
        g0.w = (unsigned int)((ga >> 32) & 0x1FFFFFFull)        // global_addr[56:32]
             | (2u << 30);                                      // type=2 (image)
        v8i g1;
        g1[0] = (int)(2u << 16);          // data_size=4B; no multicast/pad/iterate
        g1[1] = (int)(512u << 16);        // tensor_dim0[15:0]=512 (bits 63:48)
        g1[2] = (int)(20u << 16);         // tensor_dim0[31:16]=0, tensor_dim1[15:0]=20
        g1[3] = (int)(512u << 16);        // tensor_dim1[31:16]=0, tile_dim0=512
        g1[4] = (int)20u;                 // tile_dim1=20, tile_dim2=0
        g1[5] = (int)512;                 // tensor_dim0_stride[31:0]=512 (contiguous rows)
        g1[6] = 0;                        // stride0[47:32]=0, stride1[15:0]=0
        g1[7] = 0;                        // stride1[47:16]=0
        v4i gz4 = {0, 0, 0, 0};           // groups 2/3 unused (<=2D tensor)
        v8i gz8 = {0, 0, 0, 0, 0, 0, 0, 0};
        __builtin_amdgcn_tensor_load_to_lds(g0, g1, gz4, gz4, gz8, 0);
        __builtin_amdgcn_s_wait_tensorcnt((short)0);
    }
#else
    // Fallback: cooperative 128-bit staging.
    {
        const float4* gsrc = (const float4*)(wsf + COEF_OFF);
        float4* ldst = (float4*)sCoef;
#pragma unroll
        for (int k = 0; k < 8; ++k) ldst[tid + k * 256] = gsrc[tid + k * 256];
        const int4* gb = (const int4*)(wsf + BND_OFF);
        int4* lb = (int4*)sBnd;
        lb[tid]       = gb[tid];
        lb[tid + 256] = gb[tid + 256];
    }
#endif
    __syncthreads();

    int lane = tid & 31;
    int wid  = (blockIdx.x * 256 + tid) >> 5;   // global wave id
    int pix  = (wid << 4) + (lane & 15);        // 16 pixels per wave
    int i = pix >> 9, j = pix & 511;
    float px = __fadd_rn(-1.0f, __fmul_rn((float)i, 2.0f / 511.0f));
    float py = __fadd_rn(-1.0f, __fmul_rn((float)j, 2.0f / 511.0f));
    bool hi = (lane >= 16);

    // B (4x16): v0 = {K0=px | K2=1}, v1 = {K1=py | K3=0} (mirrors documented A layout)
    v2f B;
    B.x = hi ? 1.0f : px;
    B.y = hi ? 0.0f : py;

    int m = lane & 15;  // A row = function index
    // row m of group g -> triangle 4g + (m>>2), function m&3; lo lanes read (a,b), hi (c,0)
    int abase = ((m >> 2) * 16) + ((m & 3) * 4) + (hi ? 2 : 0);

    float bestZ = zmin;
    int bestIdx = -1;
    v8f czero = {};

    for (int g = 0; g < 128; ++g) {
        const float* ap = sCoef + (g << 6) + abase;
        v2f A;
        A.x = ap[0];
        A.y = ap[1];
        v8f d = __builtin_amdgcn_wmma_f32_16x16x4_f32(
            false, A, false, B, (short)0, czero, false, false);

        int tb = (g << 2) + (hi ? 2 : 0);  // first of this lane's 2 triangles
        const int* b0 = sBnd + tb * 4;
        bool c0 = (d[0] > 0.0f) & (d[1] > 0.0f) & (d[2] > 0.0f) &
                  (i >= b0[0]) & (i < b0[2]) & (j >= b0[1]) & (j < b0[3]);
        if (c0 & (d[3] >= bestZ)) { bestZ = d[3]; bestIdx = tb; }

        const int* b1 = sBnd + (tb + 1) * 4;
        bool c1 = (d[4] > 0.0f) & (d[5] > 0.0f) & (d[6] > 0.0f) &
                  (i >= b1[0]) & (i < b1[2]) & (j >= b1[1]) & (j < b1[3]);
        if (c1 & (d[7] >= bestZ)) { bestZ = d[7]; bestIdx = tb + 1; }
    }

    // Merge the two half-wave triangle streams (ties -> later index).
    float zo = __shfl_xor(bestZ, 16, 32);
    int   io = __shfl_xor(bestIdx, 16, 32);
    if ((zo > bestZ) || ((zo == bestZ) && (io > bestIdx))) { bestZ = zo; bestIdx = io; }

    if (lane < 16) {
        float4 o = make_float4(0.0f, 0.0f, 0.0f, 0.0f);
        if (bestIdx >= 0) {
            // Exact recompute in reference operation order for the winner.
            const float* tp = tris + bestIdx * 9;
            float ax = tp[0], ay = tp[1], az = tp[2];
            float bx = tp[3], by = tp[4], bz = tp[5];
            float cx = tp[6], cy = tp[7], cz = tp[8];
            float w   = (bx - ax) * (cy - ay) - (by - ay) * (cx - ax);
            float ws_ = (w >= 1e-9f) ? w : 1.0f;
            float pCB = (px - cx) * (by - cy) - (py - cy) * (bx - cx);
            float pCA = (px - ax) * (cy - ay) - (py - ay) * (cx - ax);
            float w1 = pCB / ws_;
            float w2 = pCA / ws_;
            float w3 = 1.0f - w1 - w2;
            o.x = w1 * ax + w2 * bx + w3 * cx;
            o.y = w1 * ay + w2 * by + w3 * cy;
            o.z = w1 * az + w2 * bz + w3 * cz;
            o.w = 1.0f;
        }
        ((float4*)out)[pix] = o;
    }
}

extern "C" void kernel_launch(void* const* d_in, const int* in_sizes, int n_in,
                              void* d_out, int out_size, void* d_ws, size_t ws_size,
                              hipStream_t stream) {
    const float* tris = (const float*)d_in[0];
    float* wsf = (float*)d_ws;
    float* out = (float*)d_out;
    prep_kernel<<<1, 512, 0, stream>>>(tris, wsf);
    // 262144 pixels / 16 per wave / 8 waves per block = 2048 blocks
    raster_kernel<<<2048, 256, 0, stream>>>(tris, wsf, out);
}